// SurfaceAwareReadout_27582279975197
// MI455X (gfx1250) — compile-verified
//
#include <hip/hip_runtime.h>
#include <hip/hip_bf16.h>

// Segment-softmax weighted readout, single pass over h (online softmax).
// gfx1250 / CDNA5: v_wmma_f32_16x16x4_f32 for both the h@Wq dots and the
// weighted row accumulation; GLOBAL_LOAD_ASYNC_TO_LDS_B128 double-buffered
// tile staging synchronized with s_wait_asynccnt.

#define DIM           128
#define NWAVES        3                       // 96 threads per block
#define ROWS_PER_WAVE 16
#define TILE_ROWS     (NWAVES * ROWS_PER_WAVE)  // 48
#define LDS_STRIDE    132                     // floats per tile row (padded)

#define AS1 __attribute__((address_space(1)))
#define AS3 __attribute__((address_space(3)))

typedef float v2f __attribute__((ext_vector_type(2)));
typedef float v8f __attribute__((ext_vector_type(8)));
typedef int   g4i __attribute__((vector_size(16)));   // matches builtin param type

#if defined(__HIP_DEVICE_COMPILE__)
#if !__has_builtin(__builtin_amdgcn_wmma_f32_16x16x4_f32)
#error "device pass: missing __builtin_amdgcn_wmma_f32_16x16x4_f32"
#endif
#if __has_builtin(__builtin_amdgcn_global_load_async_to_lds_b128)
#define HAVE_ASYNC_LDS 1
#endif
#endif

__device__ __forceinline__ float wshfl(float v, int src) { return __shfl(v, src, 32); }

__device__ __forceinline__ void wait_async0() {
#if defined(__HIP_DEVICE_COMPILE__)
#if __has_builtin(__builtin_amdgcn_s_wait_asynccnt)
  __builtin_amdgcn_s_wait_asynccnt(0);
#elif defined(HAVE_ASYNC_LDS)
  asm volatile("s_wait_asynccnt 0" ::: "memory");
#endif
#endif
}

// Copy one 512B row slice (float4 per lane) global -> LDS; zeros if !valid.
__device__ __forceinline__ void copy_row(const float* gsrc, float* ldst,
                                         int lane, bool valid) {
#if defined(HAVE_ASYNC_LDS)
  if (valid) {
    AS1 g4i* gp = (AS1 g4i*)(unsigned long long)(const void*)(gsrc + 4 * lane);
    AS3 g4i* lp = (AS3 g4i*)(unsigned int)(unsigned long long)(void*)(ldst + 4 * lane);
    __builtin_amdgcn_global_load_async_to_lds_b128(gp, lp, 0, 0);
  } else {
    *(float4*)(ldst + 4 * lane) = make_float4(0.f, 0.f, 0.f, 0.f);
  }
#else
  float4 vv = make_float4(0.f, 0.f, 0.f, 0.f);
  if (valid) vv = *(const float4*)(gsrc + 4 * lane);
  *(float4*)(ldst + 4 * lane) = vv;
#endif
}

__device__ __forceinline__ v8f v8f_zero() {
  v8f r;
#pragma unroll
  for (int i = 0; i < 8; ++i) r[i] = 0.0f;
  return r;
}

__device__ __forceinline__ v8f v8f_scale(v8f v, float s) {
  v8f r;
#pragma unroll
  for (int i = 0; i < 8; ++i) r[i] = v[i] * s;
  return r;
}

__global__ __launch_bounds__(NWAVES * 32) void seg_softmax_readout_kernel(
    const float* __restrict__ h,
    const float* __restrict__ wq,
    const float* __restrict__ sp,
    const float* __restrict__ ep,
    const int*   __restrict__ batch,
    const unsigned char* __restrict__ mask,
    float* __restrict__ out,
    int n)
{
  __shared__ __align__(16) float tiles[2][TILE_ROWS * LDS_STRIDE];
  __shared__ __align__(16) float swq[DIM];
  __shared__ __align__(16) float mAcc[NWAVES][DIM];
  __shared__ float mM[NWAVES];
  __shared__ float mD[NWAVES];

  const int b    = blockIdx.x;
  const int tid  = threadIdx.x;
  const int lane = tid & 31;
  const int wv   = tid >> 5;
  const int l15  = lane & 15;
  const int half = (lane >> 4) << 1;   // 0 for lanes 0-15, 2 for lanes 16-31
  const float NEG_INF = -__builtin_inff();

  // segment bounds via binary search on sorted batch ids
  int lo = 0, hi = n;
  while (lo < hi) { int mid = (lo + hi) >> 1; if (batch[mid] < b) lo = mid + 1; else hi = mid; }
  const int segStart = lo;
  hi = n;
  while (lo < hi) { int mid = (lo + hi) >> 1; if (batch[mid] < b + 1) lo = mid + 1; else hi = mid; }
  const int segEnd = lo;

  for (int c = tid; c < DIM; c += NWAVES * 32) swq[c] = wq[c];

  float runM = NEG_INF;   // running segment max (per wave)
  float runD = 0.0f;      // running denom (per wave)
  v8f acc[8];             // weighted-sum accumulators, col block t = cols [16t,16t+16)
#pragma unroll
  for (int t = 0; t < 8; ++t) acc[t] = v8f_zero();

  // this wave stages rows [16*wv, 16*wv+16) of each tile
  auto stage = [&](int tileStart, float* buf) {
    const long gbase = (long)(tileStart + ROWS_PER_WAVE * wv) * DIM;
#pragma unroll
    for (int i = 0; i < ROWS_PER_WAVE; ++i) {
      const int gr = tileStart + ROWS_PER_WAVE * wv + i;   // wave-uniform validity
      copy_row(h + gbase + (long)i * DIM,
               buf + (ROWS_PER_WAVE * wv + i) * LDS_STRIDE, lane, gr < segEnd);
    }
  };

  int buf = 0;
  if (segStart < segEnd) stage(segStart, tiles[0]);

  for (int tileStart = segStart; tileStart < segEnd; tileStart += TILE_ROWS) {
    wait_async0();        // this wave's async copies into tiles[buf] are done
    __syncthreads();      // all waves' copies landed; prior compute on tiles[buf^1] done

    if (tileStart + TILE_ROWS < segEnd)
      stage(tileStart + TILE_ROWS, tiles[buf ^ 1]);   // overlaps with compute below

    const float* tb0 = tiles[buf];

    // ---- phase 1: dots for this wave's 16 rows via WMMA 16x16x4 f32
    // A (16x4 f32): lanes 0-15 hold row=lane K=0,1; lanes 16-31 row=lane-16 K=2,3
    // B (4x16 f32): B[k][n] = Wq[4c+k] broadcast across n (same LDS addr per half)
    v8f dotAcc = v8f_zero();
    {
      const float* rowp = tb0 + (ROWS_PER_WAVE * wv + l15) * LDS_STRIDE;
#pragma unroll
      for (int c = 0; c < 32; ++c) {
        v2f a  = *(const v2f*)(rowp + 4 * c + half);
        v2f bb = *(const v2f*)(swq  + 4 * c + half);
        dotAcc = __builtin_amdgcn_wmma_f32_16x16x4_f32(false, a, false, bb,
                                                       (short)0, dotAcc, false, false);
      }
    }
    // extract dot(row r): r<8 -> D vgpr r @ lane 0; r>=8 -> D vgpr r-8 @ lane 16
    float dotv = 0.0f;
#pragma unroll
    for (int j = 0; j < 8; ++j) {
      float lov = wshfl(dotAcc[j], 0);
      float hiv = wshfl(dotAcc[j], 16);
      if (l15 == j)     dotv = lov;
      if (l15 == j + 8) dotv = hiv;
    }

    // ---- per-row logit (lane l and l+16 both hold row l15)
    const int gr = tileStart + ROWS_PER_WAVE * wv + l15;
    float x = NEG_INF;
    if (gr < segEnd) {
      const float spv = sp[gr];
      const float epv = ep[gr];
      if (mask[gr]) x = dotv + 0.5f * spv + 0.5f * epv;
    }

    // block max over the 16 rows (values duplicated across halves)
    float bm = x;
#pragma unroll
    for (int s = 16; s >= 1; s >>= 1) bm = fmaxf(bm, __shfl_xor(bm, s, 32));

    if (bm > NEG_INF) {           // wave-uniform: skip fully-masked / padded blocks
      const float newM  = fmaxf(runM, bm);
      const float scale = __expf(runM - newM);     // runM=-inf -> 0 ; unchanged -> 1
      const float e     = __expf(x - newM);        // x=-inf -> 0
      float bs = e;                                // sum within 16-lane half
#pragma unroll
      for (int s = 8; s >= 1; s >>= 1) bs += __shfl_xor(bs, s, 32);
      runD = runD * scale + bs;
      runM = newM;
      if (scale != 1.0f) {
#pragma unroll
        for (int t = 0; t < 8; ++t) acc[t] = v8f_scale(acc[t], scale);
      }

      // ---- phase 2: acc[t] += A(e-weights, 16x4) x B(h rows, 4x16) per col block
      // A[m][k] = e_{4kg+k} for all m ; B[k][n] = tile[4kg+k][16t+n]
      const float* tb = tb0 + (ROWS_PER_WAVE * wv) * LDS_STRIDE;
#pragma unroll
      for (int kg = 0; kg < 4; ++kg) {
        v2f av;
        av.x = wshfl(e, 4 * kg + half);
        av.y = wshfl(e, 4 * kg + half + 1);
        const float* r0 = tb + (4 * kg + half) * LDS_STRIDE;
        const float* r1 = r0 + LDS_STRIDE;
#pragma unroll
        for (int t = 0; t < 8; ++t) {
          v2f bv;
          bv.x = r0[16 * t + l15];
          bv.y = r1[16 * t + l15];
          acc[t] = __builtin_amdgcn_wmma_f32_16x16x4_f32(false, av, false, bv,
                                                         (short)0, acc[t], false, false);
        }
      }
    }
    buf ^= 1;
  }

  // ---- merge the waves' online-softmax states
  // acc[t] vgpr0 @ lanes 0-15 holds cols [16t+lane] (M=0 row; replicated over M)
#pragma unroll
  for (int t = 0; t < 8; ++t) {
    if (lane < 16) mAcc[wv][16 * t + l15] = acc[t][0];
  }
  if (lane == 0) { mM[wv] = runM; mD[wv] = runD; }
  __syncthreads();

  for (int c = tid; c < DIM; c += NWAVES * 32) {
    float M = NEG_INF;
#pragma unroll
    for (int w = 0; w < NWAVES; ++w) M = fmaxf(M, mM[w]);
    float A = 0.0f, D = 0.0f;
    if (M > NEG_INF) {
#pragma unroll
      for (int w = 0; w < NWAVES; ++w) {
        const float mw = mM[w];
        const float s  = (mw > NEG_INF) ? __expf(mw - M) : 0.0f;
        A += s * mAcc[w][c];
        D += s * mD[w];
      }
    }
    out[(long)b * DIM + c] = (D > 0.0f) ? (A / D) : 0.0f;
  }
}

extern "C" void kernel_launch(void* const* d_in, const int* in_sizes, int n_in,
                              void* d_out, int out_size, void* d_ws, size_t ws_size,
                              hipStream_t stream) {
  const float* h   = (const float*)d_in[0];          // [N,128] f32
  const float* wq  = (const float*)d_in[1];          // [128]  f32
  const float* sp  = (const float*)d_in[2];          // [N]    f32
  const float* ep  = (const float*)d_in[3];          // [N]    f32
  const int*   bt  = (const int*)d_in[4];            // [N]    int (sorted)
  const unsigned char* mk = (const unsigned char*)d_in[5];  // [N] bool
  const int n    = in_sizes[2];                      // N from surface_prior
  const int nseg = out_size / DIM;                   // B = 512

  seg_softmax_readout_kernel<<<dim3(nseg), dim3(NWAVES * 32), 0, stream>>>(
      h, wq, sp, ep, bt, mk, (float*)d_out, n);
}